// RPN_41317585388041
// MI455X (gfx1250) — compile-verified
//
#include <hip/hip_runtime.h>
#include <math.h>

// ---------------------------------------------------------------------------
// RPN pipeline for MI455X (gfx1250, wave32).
//  - Conv heads as implicit GEMM on V_WMMA_F32_16X16X4_F32 (exact f32: output
//    feeds sigmoid>0.5 threshold + NMS ordering, so no low-precision WMMA).
//  - 192KB LDS row-stage per workgroup (CDNA5 WGP has 320KB LDS), moved with
//    GLOBAL_LOAD_ASYNC_TO_LDS_B128 (ASYNCcnt) when the toolchain exposes it.
//  - Weights repacked once into the WMMA 4x16 B-fragment lane layout.
// ---------------------------------------------------------------------------

typedef __attribute__((ext_vector_type(2))) float v2f;
typedef __attribute__((ext_vector_type(4))) float v4f;
typedef __attribute__((ext_vector_type(8))) float v8f;
typedef __attribute__((ext_vector_type(4))) int v4i;
typedef __attribute__((address_space(1))) v4i gv4i;   // global int4
typedef __attribute__((address_space(3))) v4i lv4i;   // LDS int4

#define FH 64
#define FW 64
#define CIN 256
#define NB 4
#define NANCH 36864           // 64*64*9
#define NPAD 48               // 45 output channels padded to 3 N-tiles of 16
#define KTOT 2304             // 256 * 9 taps

#if __has_builtin(__builtin_amdgcn_global_load_async_to_lds_b128)
#define HAS_ASYNC_LDS 1
#else
#define HAS_ASYNC_LDS 0
#endif

// ---------------------------------------------------------------------------
// Kernel 0: repack conv weights into WMMA B-fragment layout.
// B matrix (4x16 f32, KxN): lanes 0-15 hold (K0,K1) for N=lane, lanes 16-31
// hold (K2,K3) for N=lane-16.  Index: [tap][c4][ntile][lane] -> float2.
// k = tap*256 + c ;  n<9 -> w_cls[n], 9<=n<45 -> w_loc[n-9], n>=45 -> 0.
// ---------------------------------------------------------------------------
__global__ void prep_weights(const float* __restrict__ wc,
                             const float* __restrict__ wl,
                             float* __restrict__ Bprep) {
  int t = blockIdx.x * blockDim.x + threadIdx.x;   // 9*64*3*32 = 55296
  if (t >= 9 * 64 * 3 * 32) return;
  int lane = t & 31;
  int rest = t >> 5;
  int nt = rest % 3; rest /= 3;
  int c4 = rest % 64;
  int tap = rest / 64;
  int n = nt * 16 + (lane & 15);
  int c0 = c4 * 4 + (lane >> 4) * 2;
  float v0 = 0.f, v1 = 0.f;
  if (n < 45) {
    if (n < 9) {
      v0 = wc[(n * CIN + c0) * 9 + tap];
      v1 = wc[(n * CIN + c0 + 1) * 9 + tap];
    } else {
      v0 = wl[((n - 9) * CIN + c0) * 9 + tap];
      v1 = wl[((n - 9) * CIN + c0 + 1) * 9 + tap];
    }
  }
  Bprep[t * 2 + 0] = v0;
  Bprep[t * 2 + 1] = v1;
}

// ---------------------------------------------------------------------------
// Kernel 1: fused cls+loc 3x3 conv via WMMA implicit GEMM.
// Grid: B*FH workgroups (one output row each), 128 threads = 4 waves.
// LDS: feature rows y-1..y+1, all 256 channels: [c][dy][x] = 192 KB (dynamic).
// Each wave: 16-wide x-strip (M), 3 N-tiles of 16, K-loop 9 taps x 64 quads.
// ---------------------------------------------------------------------------
__global__ void conv_wmma(const float* __restrict__ feat,
                          const float* __restrict__ Bprep,
                          float* __restrict__ scores,   // [B][4096][9] sigmoid
                          float* __restrict__ locbuf) { // [B][4096][9][4]
  extern __shared__ float lds[];                 // 256*3*64 floats
  const int tid = threadIdx.x;                   // 0..127
  const int b = blockIdx.x >> 6;
  const int y = blockIdx.x & 63;
  const float* fb_ = feat + (size_t)b * CIN * FH * FW;

  // Stage 3 feature rows (zero-padded in y) in 16B units.  In-range rows go
  // memory -> LDS directly via the async DMA path (no VGPR round-trip, all
  // 96 transfers per wave in flight on ASYNCcnt); pad rows are ds-stored 0.
  for (int q = tid; q < (CIN * 3 * FW) / 4; q += 128) {
    const int idx = q * 4;
    const int c = idx / 192;
    const int rem = idx - c * 192;
    const int dy = rem >> 6;
    const int x = rem & 63;
    const int yy = y + dy - 1;
    float* ldst = &lds[idx];
    if (yy >= 0 && yy < FH) {
      const float* gsrc = fb_ + (c * FH + yy) * FW + x;
#if HAS_ASYNC_LDS
      __builtin_amdgcn_global_load_async_to_lds_b128((gv4i*)gsrc, (lv4i*)ldst,
                                                     0, 0);
#else
      *(v4f*)ldst = *(const v4f*)gsrc;
#endif
    } else {
      v4f z = {0.f, 0.f, 0.f, 0.f};
      *(v4f*)ldst = z;
    }
  }
#if HAS_ASYNC_LDS
#if __has_builtin(__builtin_amdgcn_s_wait_asynccnt)
  __builtin_amdgcn_s_wait_asynccnt(0);
#else
  asm volatile("s_wait_asynccnt 0x0" ::: "memory");
#endif
#endif
  __syncthreads();

  const int wave = tid >> 5;
  const int lane = tid & 31;
  const int half = lane >> 4;                    // A layout: K-pair select
  const int xl = lane & 15;                      // A layout: M = lane&15
  const int x0 = wave * 16;
  const int x = x0 + xl;

  v8f acc0 = {0.f, 0.f, 0.f, 0.f, 0.f, 0.f, 0.f, 0.f};
  v8f acc1 = {0.f, 0.f, 0.f, 0.f, 0.f, 0.f, 0.f, 0.f};
  v8f acc2 = {0.f, 0.f, 0.f, 0.f, 0.f, 0.f, 0.f, 0.f};
  const v2f* __restrict__ Bv = (const v2f*)Bprep;

  for (int tap = 0; tap < 9; ++tap) {
    const int ky = tap / 3, kx = tap - ky * 3;
    const int xx = x + kx - 1;
    const bool inx = (xx >= 0) && (xx < FW);
    const int xxc = inx ? xx : 0;                // clamped read, masked below
    const float* lrow = lds + ky * 64 + xxc;     // + c*192 per channel
    const int bbase = (tap * 64) * 3 * 32 + lane;
#pragma unroll 4
    for (int c4 = 0; c4 < 64; ++c4) {
      const int cA = c4 * 4 + half * 2;
      float a0 = lrow[cA * 192];
      float a1 = lrow[cA * 192 + 192];
      a0 = inx ? a0 : 0.f;                       // SAME padding in x (cndmask)
      a1 = inx ? a1 : 0.f;
      v2f a = {a0, a1};
      const int bi = bbase + c4 * 3 * 32;
      v2f bf0 = Bv[bi];
      v2f bf1 = Bv[bi + 32];
      v2f bf2 = Bv[bi + 64];
      acc0 = __builtin_amdgcn_wmma_f32_16x16x4_f32(false, a, false, bf0,
                                                   (short)0, acc0, false, false);
      acc1 = __builtin_amdgcn_wmma_f32_16x16x4_f32(false, a, false, bf1,
                                                   (short)0, acc1, false, false);
      acc2 = __builtin_amdgcn_wmma_f32_16x16x4_f32(false, a, false, bf2,
                                                   (short)0, acc2, false, false);
    }
  }

  // C/D layout: VGPR r -> M = r (lanes 0-15) or 8+r (lanes 16-31); N = lane&15.
  const int rowbase = b * (FH * FW) + y * FW;
#pragma unroll
  for (int t = 0; t < 3; ++t) {
    v8f A = (t == 0) ? acc0 : (t == 1) ? acc1 : acc2;
    const int n = t * 16 + xl;
#pragma unroll
    for (int r = 0; r < 8; ++r) {
      const int xo = x0 + half * 8 + r;
      const float val = A[r];
      const int pos = rowbase + xo;
      if (n < 9) {
        scores[pos * 9 + n] = 1.f / (1.f + expf(-val));
      } else if (n < 45) {
        const int m = n - 9;
        locbuf[(pos * 9 + (m >> 2)) * 4 + (m & 3)] = val;
      }
    }
  }
}

// ---------------------------------------------------------------------------
// Kernel 2: box decode + clip + validity; swork = valid ? score : -1.
// ---------------------------------------------------------------------------
__global__ void decode_kernel(const float* __restrict__ scores,
                              const float* __restrict__ locbuf,
                              const float* __restrict__ anchors,
                              const int* __restrict__ stridep,
                              float* __restrict__ boxesbuf,
                              float* __restrict__ swork) {
  int i = blockIdx.x * blockDim.x + threadIdx.x;
  if (i >= NB * NANCH) return;
  int r = i % NANCH;
  float s = scores[i];
  const float* l = locbuf + (size_t)i * 4;
  const float* a = anchors + (size_t)r * 4;
  float aw = a[2] - a[0], ah = a[3] - a[1];
  float acx = a[0] + 0.5f * aw, acy = a[1] + 0.5f * ah;
  float cx = acx + l[0] * aw;
  float cy = acy + l[1] * ah;
  float w = aw * expf(l[2]);
  float h = ah * expf(l[3]);
  float x1 = fminf(fmaxf(cx - 0.5f * w, 0.f), 512.f);
  float y1 = fminf(fmaxf(cy - 0.5f * h, 0.f), 512.f);
  float x2 = fminf(fmaxf(cx + 0.5f * w, 0.f), 512.f);
  float y2 = fminf(fmaxf(cy + 0.5f * h, 0.f), 512.f);
  float* bo = boxesbuf + (size_t)i * 4;
  bo[0] = x1; bo[1] = y1; bo[2] = x2; bo[3] = y2;
  float ms = 2.f * (float)stridep[0];
  bool valid = (s > 0.5f) && ((x2 - x1) >= ms) && ((y2 - y1) >= ms);
  swork[i] = valid ? s : -1.f;
}

// ---------------------------------------------------------------------------
// Kernel 3: greedy NMS, 32 rounds, one 1024-thread block per batch image.
// Tie-break = lowest index (matches jnp.argmax).  Writes eval_boxes, ok,
// and integer feature-map boxes for the ROI kernel.
// ---------------------------------------------------------------------------
#define NMS_T 1024
__global__ void nms_kernel(const float* __restrict__ boxesbuf,
                           const float* __restrict__ scores,
                           float* __restrict__ swork,
                           const int* __restrict__ stridep,
                           float* __restrict__ d_out,
                           int* __restrict__ fbbuf,
                           int* __restrict__ okbuf) {
  __shared__ float sval[NMS_T];
  __shared__ int sidx[NMS_T];
  __shared__ float sbox[4];
  __shared__ int spick;
  __shared__ int pickS[32];
  __shared__ int okS[32];
  const int b = blockIdx.x, tid = threadIdx.x;
  const int N = NANCH;
  const float* bx = boxesbuf + (size_t)b * N * 4;
  float* s = swork + (size_t)b * N;

  for (int round = 0; round < 32; ++round) {
    float bv = -1e38f;
    int bi = 0;
    for (int e = tid; e < N; e += NMS_T) {      // ascending: strict > keeps first
      float v = s[e];
      if (v > bv) { bv = v; bi = e; }
    }
    sval[tid] = bv; sidx[tid] = bi;
    __syncthreads();
    for (int off = NMS_T / 2; off > 0; off >>= 1) {
      if (tid < off) {
        float v2 = sval[tid + off]; int i2 = sidx[tid + off];
        if (v2 > sval[tid] || (v2 == sval[tid] && i2 < sidx[tid])) {
          sval[tid] = v2; sidx[tid] = i2;
        }
      }
      __syncthreads();
    }
    if (tid == 0) {
      int pi = sidx[0];
      spick = pi;
      pickS[round] = pi;
      okS[round] = (sval[0] > 0.f) ? 1 : 0;
      sbox[0] = bx[pi * 4 + 0]; sbox[1] = bx[pi * 4 + 1];
      sbox[2] = bx[pi * 4 + 2]; sbox[3] = bx[pi * 4 + 3];
    }
    __syncthreads();
    const float X1 = sbox[0], Y1 = sbox[1], X2 = sbox[2], Y2 = sbox[3];
    const float area1 = (X2 - X1) * (Y2 - Y1);
    const int pi = spick;
    for (int e = tid; e < N; e += NMS_T) {
      float ex1 = bx[e * 4], ey1 = bx[e * 4 + 1];
      float ex2 = bx[e * 4 + 2], ey2 = bx[e * 4 + 3];
      float ix1 = fmaxf(X1, ex1), iy1 = fmaxf(Y1, ey1);
      float ix2 = fminf(X2, ex2), iy2 = fminf(Y2, ey2);
      float inter = fmaxf(ix2 - ix1, 0.f) * fmaxf(iy2 - iy1, 0.f);
      float a2 = (ex2 - ex1) * (ey2 - ey1);
      float iou = inter / (area1 + a2 - inter + 1e-9f);
      if (iou > 0.3f || e == pi) s[e] = -1.f;
    }
    __syncthreads();
  }

  if (tid < 32) {
    int pi = pickS[tid];
    int ok = okS[tid];
    float sc = scores[(size_t)b * N + pi];
    float X1 = bx[pi * 4], Y1 = bx[pi * 4 + 1];
    float X2 = bx[pi * 4 + 2], Y2 = bx[pi * 4 + 3];
    float* eb = d_out + ((size_t)b * 32 + tid) * 5;
    eb[0] = ok ? X1 : 0.f; eb[1] = ok ? Y1 : 0.f;
    eb[2] = ok ? X2 : 0.f; eb[3] = ok ? Y2 : 0.f;
    eb[4] = ok ? sc : 0.f;
    d_out[NB * 32 * 5 + b * 32 + tid] = ok ? 1.f : 0.f;
    int st = stridep[0];
    int ib0 = ((int)X1) / st, ib1 = ((int)Y1) / st;
    int ib2 = ((int)X2) / st, ib3 = ((int)Y2) / st;
    int fx1 = min(max(ib0, 0), FW - 1);
    int fy1 = min(max(ib1, 0), FH - 1);
    int fx2 = min(max(ib2, fx1 + 1), FW);
    int fy2 = min(max(ib3, fy1 + 1), FH);
    int* f = fbbuf + (b * 32 + tid) * 4;
    f[0] = fx1; f[1] = fy1; f[2] = fx2; f[3] = fy2;
    okbuf[b * 32 + tid] = ok;
  }
}

// ---------------------------------------------------------------------------
// Kernel 4: adaptive 7x7 max pool per ROI.  Block = one ROI, thread = channel.
// Bins: start = floor(i*h/7), end = ceil((i+1)*h/7)  (matches reference).
// ---------------------------------------------------------------------------
__global__ void roi_kernel(const float* __restrict__ feat,
                           const int* __restrict__ fbbuf,
                           const int* __restrict__ okbuf,
                           float* __restrict__ rois) {
  const int blk = blockIdx.x;                    // 0..127
  const int c = threadIdx.x;                     // 0..255
  const int b = blk >> 5;
  const int* f = fbbuf + blk * 4;
  const int fx1 = f[0], fy1 = f[1], fx2 = f[2], fy2 = f[3];
  const int ok = okbuf[blk];
  const float* fm = feat + ((size_t)(b * CIN + c)) * FH * FW;
  float* out = rois + ((size_t)blk * CIN + c) * 49;
  const int h = fy2 - fy1, w = fx2 - fx1;
  for (int i7 = 0; i7 < 7; ++i7) {
    const int rs = fy1 + (i7 * h) / 7;
    const int re = fy1 + ((i7 + 1) * h + 6) / 7;
    for (int j7 = 0; j7 < 7; ++j7) {
      const int cs = fx1 + (j7 * w) / 7;
      const int ce = fx1 + ((j7 + 1) * w + 6) / 7;
      float m = -1e30f;
      for (int rr = rs; rr < re; ++rr)
        for (int cc = cs; cc < ce; ++cc)
          m = fmaxf(m, fm[rr * FW + cc]);
      out[i7 * 7 + j7] = ok ? m : 0.f;
    }
  }
}

// ---------------------------------------------------------------------------
extern "C" void kernel_launch(void* const* d_in, const int* in_sizes, int n_in,
                              void* d_out, int out_size, void* d_ws, size_t ws_size,
                              hipStream_t stream) {
  const float* feat    = (const float*)d_in[0];   // [4,256,64,64]
  const float* anchors = (const float*)d_in[1];   // [36864,4]
  const float* wc      = (const float*)d_in[2];   // [9,256,3,3]
  const float* wl      = (const float*)d_in[3];   // [36,256,3,3]
  const int*   stridep = (const int*)d_in[4];     // [1] == 8
  float* out = (float*)d_out;

  char* ws = (char*)d_ws;
  float* Bprep    = (float*)ws; ws += (size_t)9 * 64 * 3 * 32 * 2 * 4;  // 442 KB
  float* scores   = (float*)ws; ws += (size_t)NB * NANCH * 4;           // 576 KB
  float* locbuf   = (float*)ws; ws += (size_t)NB * NANCH * 4 * 4;       // 2.25 MB
  float* boxesbuf = (float*)ws; ws += (size_t)NB * NANCH * 4 * 4;       // 2.25 MB
  float* swork    = (float*)ws; ws += (size_t)NB * NANCH * 4;           // 576 KB
  int* fbbuf      = (int*)ws;   ws += (size_t)NB * 32 * 4 * 4;
  int* okbuf      = (int*)ws;   ws += (size_t)NB * 32 * 4;

  prep_weights<<<216, 256, 0, stream>>>(wc, wl, Bprep);

  const size_t lds_bytes = (size_t)CIN * 3 * FW * sizeof(float);        // 192 KB
  conv_wmma<<<NB * FH, 128, lds_bytes, stream>>>(feat, Bprep, scores, locbuf);

  decode_kernel<<<(NB * NANCH + 255) / 256, 256, 0, stream>>>(
      scores, locbuf, anchors, stridep, boxesbuf, swork);

  nms_kernel<<<NB, NMS_T, 0, stream>>>(boxesbuf, scores, swork, stridep,
                                       out, fbbuf, okbuf);

  roi_kernel<<<NB * 32, CIN, 0, stream>>>(feat, fbbuf, okbuf,
                                          out + NB * 32 * 5 + NB * 32);
}